// UseGlobalContext_60584808678067
// MI455X (gfx1250) — compile-verified
//
#include <hip/hip_runtime.h>

typedef __attribute__((ext_vector_type(2))) float v2f;
typedef __attribute__((ext_vector_type(4))) float v4f;
typedef __attribute__((ext_vector_type(8))) float v8f;

#define NUM_GRAPHS 1024
#define IN_F 128
#define OUT_F 128
#define LDX 132   // padded LDS row stride (floats): 132 % 64 == 4 -> conflict-free b64 frags, 16B aligned rows

// ---------------------------------------------------------------------------
// Kernel 1: segment mean over sorted batch_id. One block per graph, 128 threads
// (one per feature column). Binary search the [start,end) row range, then each
// thread sums its column (coalesced 512B row reads).
// ---------------------------------------------------------------------------
__global__ __launch_bounds__(128) void pool_mean_kernel(const float* __restrict__ x,
                                                        const int* __restrict__ bid,
                                                        int N, float* __restrict__ mean) {
    const int g = blockIdx.x;
    // lower_bound(bid, g)
    int lo = 0, hi = N;
    while (lo < hi) { int mid = (lo + hi) >> 1; if (bid[mid] < g) lo = mid + 1; else hi = mid; }
    const int start = lo;
    hi = N;
    while (lo < hi) { int mid = (lo + hi) >> 1; if (bid[mid] < g + 1) lo = mid + 1; else hi = mid; }
    const int end = lo;

    float s = 0.0f;
    const int c = threadIdx.x;
    for (int i = start; i < end; ++i) s += x[(long)i * IN_F + c];
    const float cnt = (float)(end - start);
    mean[g * IN_F + c] = s / fmaxf(cnt, 1.0f);
}

// ---------------------------------------------------------------------------
// Kernel 2: gctx[g][o] = b[o] + sum_k mean[g][k] * W[o][128+k]. Tiny (33 MFLOP).
// ---------------------------------------------------------------------------
__global__ __launch_bounds__(128) void graph_ctx_kernel(const float* __restrict__ mean,
                                                        const float* __restrict__ W,
                                                        const float* __restrict__ b,
                                                        float* __restrict__ gctx) {
    __shared__ float ms[IN_F];
    const int g = blockIdx.x;
    const int o = threadIdx.x;
    ms[o] = mean[g * IN_F + o];
    __syncthreads();
    float s = b[o];
    const float* w2 = W + o * 256 + 128;
#pragma unroll 8
    for (int k = 0; k < IN_F; ++k) s += ms[k] * w2[k];
    gctx[g * OUT_F + o] = s;
}

// ---------------------------------------------------------------------------
// Kernel 3: out[i] = x[i] @ W1^T + gctx[batch_id[i]]
// 128x128 output tile per block (256 threads = 8 waves, wave32).
// Wave w computes rows [16w,16w+16) x all 128 output cols via
// V_WMMA_F32_16X16X4_F32, K-loop of 32 steps over IN_F=128.
// Accumulators initialized from gctx (per-row gather), stored directly.
// ---------------------------------------------------------------------------
__global__ __launch_bounds__(256) void fused_gemm_kernel(const float* __restrict__ x,
                                                         const int* __restrict__ bid,
                                                         const float* __restrict__ W,
                                                         const float* __restrict__ gctx,
                                                         float* __restrict__ out, int N) {
    __shared__ float xs[128 * LDX];  // x tile   [m][k]
    __shared__ float ws[128 * LDX];  // W1       [n][k]

    const int tid  = threadIdx.x;
    const int wave = tid >> 5;
    const int lane = tid & 31;
    const int l15  = lane & 15;
    const int kh   = lane >> 4;                 // 0: K pair {0,1}, 1: K pair {2,3}
    const long row0 = (long)blockIdx.x * 128;

    // Stage W1 (W[n][0:128]) into LDS, row stride LDX. 4096 float4 / 256 thr.
#pragma unroll
    for (int i = tid; i < 128 * 32; i += 256) {
        const int n  = i >> 5;
        const int k4 = (i & 31) * 4;
        *(v4f*)(&ws[n * LDX + k4]) = *(const v4f*)(W + n * 256 + k4);
    }
    // Stage x tile (zero-fill OOB rows so WMMA runs with full EXEC).
#pragma unroll
    for (int i = tid; i < 128 * 32; i += 256) {
        const int m  = i >> 5;
        const int k4 = (i & 31) * 4;
        const long r = row0 + m;
        v4f v = (r < N) ? *(const v4f*)(x + r * IN_F + k4) : (v4f){0.f, 0.f, 0.f, 0.f};
        *(v4f*)(&xs[m * LDX + k4]) = v;
    }
    __syncthreads();

    // Accumulator layout (16x16 f32 C/D): VGPR r -> M = r + 8*kh ; N = 16*t + l15
    // This lane's 8 rows:
    int bids[8];
#pragma unroll
    for (int r = 0; r < 8; ++r) {
        const long m = row0 + 16 * wave + 8 * kh + r;
        bids[r] = (m < N) ? bid[m] : 0;
    }

    v8f acc[8];
#pragma unroll
    for (int t = 0; t < 8; ++t) {
#pragma unroll
        for (int r = 0; r < 8; ++r)
            acc[t][r] = gctx[bids[r] * OUT_F + 16 * t + l15];
    }

    // K loop: A frag = xs[16w + l15][k0 + 2*kh + {0,1}]
    //         B frag = ws[16t + l15][k0 + 2*kh + {0,1}]
    const float* arow = &xs[(16 * wave + l15) * LDX + 2 * kh];
#pragma unroll 4
    for (int k0 = 0; k0 < IN_F; k0 += 4) {
        const v2f a = *(const v2f*)(arow + k0);
#pragma unroll
        for (int t = 0; t < 8; ++t) {
            const v2f bf = *(const v2f*)(&ws[(16 * t + l15) * LDX + k0 + 2 * kh]);
            acc[t] = __builtin_amdgcn_wmma_f32_16x16x4_f32(
                false, a, false, bf, (short)0, acc[t], false, false);
        }
    }

    // Epilogue: direct store (gctx + b already folded in).
#pragma unroll
    for (int r = 0; r < 8; ++r) {
        const long m = row0 + 16 * wave + 8 * kh + r;
        if (m < N) {
#pragma unroll
            for (int t = 0; t < 8; ++t)
                out[m * OUT_F + 16 * t + l15] = acc[t][r];
        }
    }
}

extern "C" void kernel_launch(void* const* d_in, const int* in_sizes, int n_in,
                              void* d_out, int out_size, void* d_ws, size_t ws_size,
                              hipStream_t stream) {
    const float* x   = (const float*)d_in[0];
    const int*   bid = (const int*)d_in[1];
    const float* W   = (const float*)d_in[2];  // [128, 256]
    const float* b   = (const float*)d_in[3];  // [128]
    float*       out = (float*)d_out;
    const int N = in_sizes[0] / IN_F;

    float* mean = (float*)d_ws;                 // [1024, 128]
    float* gctx = mean + NUM_GRAPHS * IN_F;     // [1024, 128]

    pool_mean_kernel<<<NUM_GRAPHS, 128, 0, stream>>>(x, bid, N, mean);
    graph_ctx_kernel<<<NUM_GRAPHS, 128, 0, stream>>>(mean, W, b, gctx);

    const int nblocks = (N + 127) / 128;
    fused_gemm_kernel<<<nblocks, 256, 0, stream>>>(x, bid, W, gctx, out, N);
}